// GraphFeatureExtractor_38585986187613
// MI455X (gfx1250) — compile-verified
//
#include <hip/hip_runtime.h>
#include <hip/hip_bf16.h>

#define N_NODES 50000
#define N_EDGES 800000
#define NFEAT   128
#define HID     256
#define NGRAPH  512
#define BN_EPS  1e-5f

typedef float v2f __attribute__((ext_vector_type(2)));
typedef float v8f __attribute__((ext_vector_type(8)));

__device__ __forceinline__ void atomicAddF(float* p, float v) {
    __hip_atomic_fetch_add(p, v, __ATOMIC_RELAXED, __HIP_MEMORY_SCOPE_AGENT);
}

// ---------------------------------------------------------------- utilities
__global__ void zero_f(float* __restrict__ p, int n) {
    int i = blockIdx.x * blockDim.x + threadIdx.x;
    if (i < n) p[i] = 0.0f;
}

// Z[row*256 + f] = x[row*128 + f]   (layer-1 init: z starts as h)
__global__ void copy128to256(float* __restrict__ z, const float* __restrict__ x) {
    int i = blockIdx.x * blockDim.x + threadIdx.x;   // over 50000*128
    int row = i >> 7, f = i & 127;
    z[row * HID + f] = x[i];
}

// ------------------------------------------------------- edge aggregation
// one block per edge, blockDim = feature count (128 or 256)
__global__ void edge_agg(float* __restrict__ z, const float* __restrict__ h,
                         const int* __restrict__ src, const int* __restrict__ dst,
                         int ldz, int ldh) {
    int e = blockIdx.x;
    int f = threadIdx.x;
    int s = src[e];
    int d = dst[e];
    atomicAddF(&z[(long)d * ldz + f], h[(long)s * ldh + f]);
}

// ------------------------------------------------------- WMMA f32 GEMM
// out[m,n] = relu(sum_k in[m,k] * w[k,n] + bias[n]),  m in 16-row strip per block
// in has row stride 256 (first K columns valid), out has row stride 256.
// 16 waves per block; wave w computes the 16x16 tile at columns [16w,16w+16).
// Safe in-place (out==in): the block's 16 input rows are fully staged in LDS
// before any write.
template <int K>
__global__ __launch_bounds__(512)
void gemm_relu(const float* __restrict__ in, const float* __restrict__ w,
               const float* __restrict__ bias, float* __restrict__ out) {
    constexpr int LDA = K + 4;           // pad: (row*LDA) % 64 banks = 4*row -> conflict-free
    __shared__ float As[16 * LDA];

    const int m0 = blockIdx.x << 4;

    // cooperative stage of the 16 x K A-tile
    for (int i = threadIdx.x; i < 16 * K; i += 512) {
        int r = i / K, c = i % K;
        As[r * LDA + c] = in[(long)(m0 + r) * HID + c];
    }
    __syncthreads();

    const int lane = threadIdx.x & 31;
    const int wave = threadIdx.x >> 5;   // 0..15
    const int n0   = wave << 4;
    const int half = lane >> 4;          // 0: K,K+1   1: K+2,K+3
    const int l15  = lane & 15;

    v8f acc = {0.f, 0.f, 0.f, 0.f, 0.f, 0.f, 0.f, 0.f};

#pragma unroll 4
    for (int k = 0; k < K; k += 4) {
        // A fragment (16x4 f32): lanes 0-15 rows 0-15 K..K+1, lanes 16-31 rows 0-15 K+2..K+3
        const float* ap = &As[l15 * LDA + k + (half << 1)];
        v2f a;
        a.x = ap[0];
        a.y = ap[1];
        // B fragment (4x16 f32): lane holds w[k + 2*half + {0,1}][n0 + l15]
        const float* bp = &w[(long)(k + (half << 1)) * HID + n0 + l15];
        v2f b;
        b.x = bp[0];
        b.y = bp[HID];
        acc = __builtin_amdgcn_wmma_f32_16x16x4_f32(false, a, false, b,
                                                    (short)0, acc, false, false);
    }

    // D layout: VGPR r, lanes 0-15 -> row m0+r col n0+lane; lanes 16-31 -> row m0+r+8
    const int col  = n0 + l15;
    const float bv = bias[col];
#pragma unroll
    for (int r = 0; r < 8; ++r) {
        int row = m0 + r + (half << 3);
        float v = acc[r] + bv;
        out[(long)row * HID + col] = v > 0.f ? v : 0.f;
    }
}

// ------------------------------------------------------- BatchNorm
// partial column sums / sumsq, coalesced rows, atomics into stats[0..255]=sum,
// stats[256..511]=sumsq
__global__ void bn_stats(const float* __restrict__ h, float* __restrict__ stats,
                         int rows_per_block) {
    int f  = threadIdx.x;                // 256
    int r0 = blockIdx.x * rows_per_block;
    float s = 0.f, ss = 0.f;
    for (int r = 0; r < rows_per_block; ++r) {
        float v = h[(long)(r0 + r) * HID + f];
        s += v;
        ss += v * v;
    }
    atomicAddF(&stats[f], s);
    atomicAddF(&stats[HID + f], ss);
}

// normalize in place; optionally also write the next layer's z initializer
__global__ void bn_apply(float* __restrict__ h, const float* __restrict__ stats,
                         const float* __restrict__ gamma, const float* __restrict__ beta,
                         float* __restrict__ zout) {
    long i = (long)blockIdx.x * blockDim.x + threadIdx.x;  // over 50000*256
    int f = (int)(i & (HID - 1));
    const float invN = 1.0f / (float)N_NODES;
    float mu  = stats[f] * invN;
    float var = stats[HID + f] * invN - mu * mu;
    float inv = rsqrtf(var + BN_EPS);
    float v = gamma[f] * (h[i] - mu) * inv + beta[f];
    h[i] = v;
    if (zout) zout[i] = v;
}

// ------------------------------------------------------- pooling
__global__ void pool_sum(const float* __restrict__ h, const int* __restrict__ batch,
                         float* __restrict__ out, float* __restrict__ counts) {
    int n = blockIdx.x;
    int f = threadIdx.x;
    int g = batch[n];
    atomicAddF(&out[(long)g * HID + f], h[(long)n * HID + f]);
    if (f == 0) atomicAddF(&counts[g], 1.0f);
}

__global__ void pool_div(float* __restrict__ out, const float* __restrict__ counts) {
    int i = blockIdx.x * blockDim.x + threadIdx.x;   // over 512*256
    int g = i >> 8;
    out[i] = out[i] / fmaxf(counts[g], 1.0f);
}

// ======================================================================
extern "C" void kernel_launch(void* const* d_in, const int* in_sizes, int n_in,
                              void* d_out, int out_size, void* d_ws, size_t ws_size,
                              hipStream_t stream) {
    const float* x   = (const float*)d_in[0];
    const int*   ei  = (const int*)d_in[1];
    const int*   bat = (const int*)d_in[2];
    const float* w1a = (const float*)d_in[3];
    const float* b1a = (const float*)d_in[4];
    const float* w2a = (const float*)d_in[5];
    const float* b2a = (const float*)d_in[6];
    const float* ga  = (const float*)d_in[7];
    const float* ba  = (const float*)d_in[8];
    const float* w1s = (const float*)d_in[9];
    const float* b1s = (const float*)d_in[10];
    const float* w2s = (const float*)d_in[11];
    const float* b2s = (const float*)d_in[12];
    const float* gs  = (const float*)d_in[13];
    const float* bs  = (const float*)d_in[14];

    const int* src = ei;
    const int* dst = ei + N_EDGES;

    float* W      = (float*)d_ws;
    float* Z      = W;                                  // 50000*256
    float* A      = W + (long)N_NODES * HID;            // 50000*256
    float* stats  = A + (long)N_NODES * HID;            // 512
    float* counts = stats + 2 * HID;                    // 512
    float* out    = (float*)d_out;                      // 512*256

    const int NE = (int)((long)N_NODES * HID);          // 12.8M
    const int GM = N_NODES / 16;                        // 3125 row strips

    // ---------------- layer 1 (feat 128 -> 256) ----------------
    copy128to256<<<(N_NODES * NFEAT) / 256, 256, 0, stream>>>(Z, x);
    edge_agg<<<N_EDGES, NFEAT, 0, stream>>>(Z, x, src, dst, HID, NFEAT);
    gemm_relu<128><<<GM, 512, 0, stream>>>(Z, w1a, b1a, Z);        // in place
    gemm_relu<256><<<GM, 512, 0, stream>>>(Z, w2a, b2a, A);
    zero_f<<<2, 256, 0, stream>>>(stats, 2 * HID);
    bn_stats<<<400, 256, 0, stream>>>(A, stats, 125);
    bn_apply<<<NE / 256, 256, 0, stream>>>(A, stats, ga, ba, Z);

    // ---------------- layers 2 & 3 (256 -> 256) ----------------
    for (int l = 0; l < 2; ++l) {
        const float* w1 = w1s + (long)l * HID * HID;
        const float* b1 = b1s + (long)l * HID;
        const float* w2 = w2s + (long)l * HID * HID;
        const float* b2 = b2s + (long)l * HID;
        const float* g  = gs + (long)l * HID;
        const float* b  = bs + (long)l * HID;
        edge_agg<<<N_EDGES, HID, 0, stream>>>(Z, A, src, dst, HID, HID);
        gemm_relu<256><<<GM, 512, 0, stream>>>(Z, w1, b1, Z);      // in place
        gemm_relu<256><<<GM, 512, 0, stream>>>(Z, w2, b2, A);
        zero_f<<<2, 256, 0, stream>>>(stats, 2 * HID);
        bn_stats<<<400, 256, 0, stream>>>(A, stats, 125);
        bn_apply<<<NE / 256, 256, 0, stream>>>(A, stats, g, b,
                                               (l == 0) ? Z : nullptr);
    }

    // ---------------- global mean pool ----------------
    zero_f<<<(NGRAPH * HID) / 256, 256, 0, stream>>>(out, NGRAPH * HID);
    zero_f<<<2, 256, 0, stream>>>(counts, NGRAPH);
    pool_sum<<<N_NODES, HID, 0, stream>>>(A, bat, out, counts);
    pool_div<<<(NGRAPH * HID) / 256, 256, 0, stream>>>(out, counts);
}